// AnchorTargetLayer_26749056320271
// MI455X (gfx1250) — compile-verified
//
#include <hip/hip_runtime.h>
#include <cstdint>

#define N_ANCHORS 100000
#define BATCH     4
#define N_GT      64
#define BLOCK     256           // 8 wave32 per block
#define APT       2             // anchors per thread (two independent VOPD-friendly chains)

__device__ __forceinline__ void iou_step(const float4& a, float a_area,
                                         const float4& g, float g_area,
                                         float& best) {
    float ix1 = fmaxf(a.x, g.x);
    float iy1 = fmaxf(a.y, g.y);
    float ix2 = fminf(a.z, g.z);
    float iy2 = fminf(a.w, g.w);
    float iw  = fmaxf(ix2 - ix1, 0.0f);
    float ih  = fmaxf(iy2 - iy1, 0.0f);
    float inter = iw * ih;
    // union >= max(a_area, g_area) > 0 always (box wh in [16,256]), and
    // inter==0 -> 0/union == 0, so the reference's where() guard is redundant.
    float uni = (a_area + g_area) - inter;
    best = fmaxf(best, __fdividef(inter, uni));   // v_rcp_f32 + v_mul_f32
}

__global__ __launch_bounds__(BLOCK)
void AnchorTargetLayer_26749056320271_kernel(const float* __restrict__ anchors,
                                             const float* __restrict__ gt,
                                             float* __restrict__ out) {
    // One batch's GT table per block: 64 boxes (1 KB) + per-box area.
    __shared__ float4 s_box[N_GT];
    __shared__ float  s_area[N_GT];

    const int tid = threadIdx.x;
    const int b   = blockIdx.y;             // batch handled by this block

    // --- Stage this batch's GT boxes via CDNA5 async global->LDS DMA ------
    // Waves 0-1 (tid 0..63) each move one float4; tracked by ASYNCcnt.
    if (tid < N_GT) {
        unsigned lds_dst = (unsigned)(uintptr_t)(&s_box[tid]);   // LDS byte address
        const float* gsrc = gt + (b * N_GT + tid) * 4;
        asm volatile("global_load_async_to_lds_b128 %0, %1, off"
                     :
                     : "v"(lds_dst), "v"(gsrc)
                     : "memory");
    }

    // --- Overlap: pull this thread's two anchors while the DMA flies ------
    const int p  = blockIdx.x * BLOCK + tid;    // anchor-pair index
    const int i0 = p * APT;
    float4 a0, a1;
    float area0 = 0.0f, area1 = 0.0f;
    const bool valid = (i0 < N_ANCHORS);        // N_ANCHORS even: pair valid together
    if (valid) {
        const float4* ap = (const float4*)anchors;
        a0 = ap[i0];                            // two coalesced global_load_b128
        a1 = ap[i0 + 1];
        area0 = (a0.z - a0.x) * (a0.w - a0.y);
        area1 = (a1.z - a1.x) * (a1.w - a1.y);
    }

    asm volatile("s_wait_asynccnt 0" ::: "memory");
    __syncthreads();

    // Precompute GT areas once per block.
    if (tid < N_GT) {
        float4 g = s_box[tid];
        s_area[tid] = (g.z - g.x) * (g.w - g.y);
    }
    __syncthreads();

    if (!valid) return;

    // --- 64 GT boxes, LDS-broadcast reads, two independent IoU chains -----
    float best0 = 0.0f, best1 = 0.0f;
    #pragma unroll 8
    for (int j = 0; j < N_GT; ++j) {
        float4 g  = s_box[j];                   // wave-uniform -> broadcast ds_load_b128
        float  ga = s_area[j];                  // merged into wide ds loads by unroll
        iou_step(a0, area0, g, ga, best0);
        iou_step(a1, area1, g, ga, best1);      // independent chain -> VOPD dual-issue
    }

    // Two consecutive outputs per lane -> one coalesced global_store_b64.
    float2 r;
    r.x = best0;
    r.y = best1;
    ((float2*)(out + (size_t)b * N_ANCHORS))[p] = r;
}

extern "C" void kernel_launch(void* const* d_in, const int* in_sizes, int n_in,
                              void* d_out, int out_size, void* d_ws, size_t ws_size,
                              hipStream_t stream) {
    const float* anchors = (const float*)d_in[0];   // (100000, 4) f32
    const float* gt      = (const float*)d_in[1];   // (4, 64, 4) f32
    float*       out     = (float*)d_out;           // (4, 100000) f32

    dim3 grid((N_ANCHORS / APT + BLOCK - 1) / BLOCK, BATCH);   // (196, 4)
    AnchorTargetLayer_26749056320271_kernel<<<grid, BLOCK, 0, stream>>>(anchors, gt, out);
}